// Slots_80083960201272
// MI455X (gfx1250) — compile-verified
//
#include <hip/hip_runtime.h>

typedef float v2f __attribute__((ext_vector_type(2)));
typedef float v8f __attribute__((ext_vector_type(8)));

#define D_SZ 512
#define S_SZ 126
#define HW_SZ 784
#define LDSR 520   // padded d-stride for the 16x512 LDS K-tile (bank-conflict control)

__global__ __launch_bounds__(512) void slots_rect_wmma(
    const float* __restrict__ feats,   // (B, D, 28, 28) f32
    const float* __restrict__ masks,   // (126, 28, 28) f32
    float* __restrict__ out)           // (B, 126, D) f32
{
  __shared__ float sR[16 * LDSR];      // K x D tile of rect sums (rows 9..15 zero)
  __shared__ float sacc[16][3][28];    // per-wave column band sums

  const int b    = blockIdx.x;
  const int tid  = threadIdx.x;
  const int wave = tid >> 5;           // 0..15
  const int lane = tid & 31;

  const float* __restrict__ fb = feats + (size_t)b * (D_SZ * HW_SZ);

  // ---------------- Phase 1: stream feats, build 9 rectangle sums per d ----
  for (int it = 0; it < D_SZ / 16; ++it) {
    const int d = it * 16 + wave;                       // 16 adjacent rows in flight
    const float* __restrict__ rowp = fb + (size_t)d * HW_SZ;
    if (lane < 28) {
      float a0 = 0.f, a1 = 0.f, a2 = 0.f;               // h-bands [0,9) [9,19) [19,28)
      #pragma unroll
      for (int h = 0; h < 28; ++h) {
        float v = rowp[h * 28 + lane];
        if (h < 9)       a0 += v;
        else if (h < 19) a1 += v;
        else             a2 += v;
      }
      sacc[wave][0][lane] = a0;
      sacc[wave][1][lane] = a1;
      sacc[wave][2][lane] = a2;
    }
    __syncthreads();
    if (lane < 16) {                                    // 9 rect sums + 7 zero pad rows
      float s = 0.f;
      if (lane < 9) {
        const int hb = lane / 3;
        const int wb = lane - hb * 3;
        const int w0 = wb * 9 + (wb >> 1);              // 0, 9, 19
        const int w1 = (wb == 2) ? 28 : (wb * 9 + 9 + (wb >> 1));
        for (int w = w0; w < w1; ++w) s += sacc[wave][hb][w];
      }
      sR[lane * LDSR + d] = s;
    }
    __syncthreads();                                    // also phase boundary on last iter
  }

  // ---------------- Phase 2: Out_b(128x512) = W(128x16) * sR(16x512) -------
  // W[s][r] = masks[s, rep(r)] (exact normalization), rows s>=126 / k>=9 are 0.
  const int tm   = wave & 7;            // M tile (s = tm*16 .. +15)
  const int tn0  = wave >> 3;           // N tile group (0 or 1)
  const int hi   = lane >> 4;           // lane half
  const int ln   = lane & 15;
  const int srow = tm * 16 + ln;

  // A fragment layout (16x4 f32): lanes 0-15 hold K = k0, k0+1; lanes 16-31 K = k0+2, k0+3
  v2f afrag[4];
  #pragma unroll
  for (int kk = 0; kk < 4; ++kk) {
    const int k0 = kk * 4 + hi * 2;
    float ax = 0.f, ay = 0.f;
    if (srow < S_SZ) {
      if (k0 < 9) {
        const int i = k0 / 3, j = k0 - i * 3;
        ax = masks[(size_t)srow * HW_SZ + (size_t)((i * 9 + (i >> 1)) * 28 + (j * 9 + (j >> 1)))];
      }
      if (k0 + 1 < 9) {
        const int k1 = k0 + 1;
        const int i = k1 / 3, j = k1 - i * 3;
        ay = masks[(size_t)srow * HW_SZ + (size_t)((i * 9 + (i >> 1)) * 28 + (j * 9 + (j >> 1)))];
      }
    }
    afrag[kk].x = ax;
    afrag[kk].y = ay;
  }

  float* __restrict__ ob = out + (size_t)b * (S_SZ * D_SZ);

  for (int t = 0; t < 16; ++t) {
    const int d0 = (tn0 * 16 + t) * 16;

    // B fragment (4x16 f32): VGPR0 lanes0-15 K=k0, lanes16-31 K=k0+2; VGPR1 K=k0+1 / k0+3
    v2f bf[4];
    #pragma unroll
    for (int kk = 0; kk < 4; ++kk) {
      const int k0 = kk * 4 + hi * 2;
      v2f bv;
      bv.x = sR[k0 * LDSR + d0 + ln];
      bv.y = sR[(k0 + 1) * LDSR + d0 + ln];
      bf[kk] = bv;
    }

    v8f c = {0.f, 0.f, 0.f, 0.f, 0.f, 0.f, 0.f, 0.f};
    #pragma unroll
    for (int kk = 0; kk < 4; ++kk) {
      c = __builtin_amdgcn_wmma_f32_16x16x4_f32(
            /*neg_a=*/false, afrag[kk],
            /*neg_b=*/false, bf[kk],
            /*c_mod=*/(short)0, c,
            /*reuse_a=*/false, /*reuse_b=*/false);
    }

    // C/D layout: VGPR v -> M = tm*16 + v + hi*8, N = ln
    #pragma unroll
    for (int v = 0; v < 8; ++v) {
      const int s = tm * 16 + hi * 8 + v;
      if (s < S_SZ) ob[(size_t)s * D_SZ + d0 + ln] = c[v];
    }
  }
}

extern "C" void kernel_launch(void* const* d_in, const int* in_sizes, int n_in,
                              void* d_out, int out_size, void* d_ws, size_t ws_size,
                              hipStream_t stream) {
  const float* feats = (const float*)d_in[0];
  const float* masks = (const float*)d_in[1];
  float* out = (float*)d_out;
  const int B = in_sizes[0] / (D_SZ * HW_SZ);   // 256
  slots_rect_wmma<<<dim3(B, 1, 1), dim3(512, 1, 1), 0, stream>>>(feats, masks, out);
}